// RGCN_62801011802251
// MI455X (gfx1250) — compile-verified
//
#include <hip/hip_runtime.h>
#include <hip/hip_bf16.h>

typedef __attribute__((ext_vector_type(2))) float v2f;
typedef __attribute__((ext_vector_type(8))) float v8f;

#define NN 7
#define HH 16
#define RR 64
#define CC 8
#define K1 (RR * NN)        // 448 : shared K-dim of both big GEMMs
#define NR (NN * RR)        // 448 : (dst, rel) groups
#define NRN (NN * RR * NN)  // 3136: (dst, rel, src) bins
#define MPAD 16             // padded M (node) dimension for WMMA fragments

// ---------------- kernel 0: zero the global (dst,rel,src) histogram ----------------
__global__ void rgcn_zero(unsigned int* __restrict__ cnt3) {
    for (int i = threadIdx.x; i < NRN; i += blockDim.x) cnt3[i] = 0u;
}

// ---------------- kernel 1: (dst,rel,src) histogram over E edges -------------------
// Memory-bound: 12 B/edge (96 MB total -> ~4 us at 23.3 TB/s). LDS-privatized per
// block (ds_add_u32), single global atomic flush of 3136 bins at the end.
__global__ void __launch_bounds__(256) rgcn_hist(const int* __restrict__ ei,
                                                 const int* __restrict__ et,
                                                 unsigned int* __restrict__ cnt3,
                                                 int E) {
    __shared__ unsigned int lh[NRN];
    for (int i = threadIdx.x; i < NRN; i += blockDim.x) lh[i] = 0u;
    __syncthreads();

    const int4* s4 = (const int4*)ei;        // src  = edge_index[0, :]
    const int4* d4 = (const int4*)(ei + E);  // dst  = edge_index[1, :]
    const int4* t4 = (const int4*)et;
    const int nq = E >> 2;
    const int stride = gridDim.x * blockDim.x;
    for (int q = blockIdx.x * blockDim.x + threadIdx.x; q < nq; q += stride) {
        int4 s = s4[q], d = d4[q], t = t4[q];
        atomicAdd(&lh[(d.x * RR + t.x) * NN + s.x], 1u);
        atomicAdd(&lh[(d.y * RR + t.y) * NN + s.y], 1u);
        atomicAdd(&lh[(d.z * RR + t.z) * NN + s.z], 1u);
        atomicAdd(&lh[(d.w * RR + t.w) * NN + s.w], 1u);
    }
    // tail (E not multiple of 4)
    if (blockIdx.x == 0 && threadIdx.x == 0) {
        for (int e = nq << 2; e < E; ++e)
            atomicAdd(&lh[(ei[E + e] * RR + et[e]) * NN + ei[e]], 1u);
    }
    __syncthreads();
    for (int i = threadIdx.x; i < NRN; i += blockDim.x) {
        unsigned int v = lh[i];
        if (v) atomicAdd(&cnt3[i], v);
    }
}

// ---------------- kernel 2: finalize (one wave32, all-WMMA math) -------------------
// All fragment sources are zero-padded in LDS so every WMMA operand load is
// unconditional (no EXEC masking in the K loops).
__global__ void __launch_bounds__(32)
rgcn_finalize(const unsigned int* __restrict__ cnt3,
              const float* __restrict__ W1, const float* __restrict__ root1,
              const float* __restrict__ b1, const float* __restrict__ W2,
              const float* __restrict__ root2, const float* __restrict__ b2,
              float* __restrict__ out) {
    __shared__ float ldsInv[NR];          // 1/max(count[dst,rel],1)
    __shared__ float ldsA[MPAD * K1];     // A[m][k]=cnt3[m,r,s]*inv (rows 7..15 = 0)
    __shared__ float ldsH[MPAD * HH];     // layer-1 activations (rows 7..15 = 0)
    __shared__ float ldsHR[K1 * MPAD];    // hr[k=(r*7+s)][c], cols 8..15 = 0
    __shared__ float ldsOut[NN * CC];

    const int tid = threadIdx.x;

    // per-(dst,rel) inverse mean normalizer
    for (int i = tid; i < NR; i += 32) {
        unsigned int s = 0;
        for (int j = 0; j < NN; ++j) s += cnt3[i * NN + j];
        ldsInv[i] = 1.0f / fmaxf((float)s, 1.0f);
    }
    __syncthreads();
    // A matrix, zero-padded to 16 rows (shared by both big GEMMs)
    for (int m = 0; m < MPAD; ++m)
        for (int k = tid; k < K1; k += 32)
            ldsA[m * K1 + k] =
                (m < NN) ? (float)cnt3[m * K1 + k] * ldsInv[m * RR + k / NN] : 0.0f;
    __syncthreads();

    // wave32 WMMA fragment coordinates (V_WMMA_F32_16X16X4_F32)
    const int half = tid >> 4;   // 0: lanes 0-15, 1: lanes 16-31
    const int col  = tid & 15;   // A: row M ; B/C/D: column N
    const int kb   = 2 * half;   // K offset held by this half-wave
    const int c8   = col & 7;    // clamped C index for W2 address padding

    // ---- GEMM1: h_pre(16x16) = A(16x448) @ W1(448x16) ----
    v8f acc = {};
#pragma unroll 4
    for (int k0 = 0; k0 < K1; k0 += 4) {
        const int kA = k0 + kb;
        v2f a, b;
        a.x = ldsA[col * K1 + kA];        // adjacent pair -> ds_load_b64
        a.y = ldsA[col * K1 + kA + 1];
        b.x = W1[kA * HH + col];
        b.y = W1[(kA + 1) * HH + col];
        acc = __builtin_amdgcn_wmma_f32_16x16x4_f32(false, a, false, b,
                                                    (short)0, acc, false, false);
    }
    // h = relu(h_pre + root1 + b1); rows 7..15 stored as exact zeros (unconditional)
#pragma unroll
    for (int v = 0; v < 8; ++v) {
        int row = v + 8 * half;
        float val =
            (row < NN) ? fmaxf(acc[v] + root1[row * HH + col] + b1[col], 0.0f) : 0.0f;
        ldsH[row * HH + col] = val;
    }
    __syncthreads();

    // ---- per-relation hr = h(7x16) @ W2[r](16x8), via WMMA ----
    // B col padding via value select (v_cndmask), never via EXEC-masked loads.
    for (int r = 0; r < RR; ++r) {
        v8f a2 = {};
#pragma unroll
        for (int kk = 0; kk < HH; kk += 4) {
            const int kA = kk + kb;
            v2f a, b;
            a.x = ldsH[col * HH + kA];    // adjacent pair -> ds_load_b64
            a.y = ldsH[col * HH + kA + 1];
            float w0 = W2[(r * HH + kA) * CC + c8];
            float w1 = W2[(r * HH + kA + 1) * CC + c8];
            b.x = (col < CC) ? w0 : 0.0f;
            b.y = (col < CC) ? w1 : 0.0f;
            a2 = __builtin_amdgcn_wmma_f32_16x16x4_f32(false, a, false, b,
                                                       (short)0, a2, false, false);
        }
        // cols 8..15 of D are exact zeros (B was zero there) -> store all 16 cols
#pragma unroll
        for (int v = 0; v < 8; ++v) {
            int s = v + 8 * half;
            if (s < NN) ldsHR[(r * NN + s) * MPAD + col] = a2[v];
        }
    }
    __syncthreads();

    // ---- GEMM2: out_pre(16x16, cols<8 valid) = A(16x448) @ HR(448x16 padded) ----
    v8f acc3 = {};
#pragma unroll 4
    for (int k0 = 0; k0 < K1; k0 += 4) {
        const int kA = k0 + kb;
        v2f a, b;
        a.x = ldsA[col * K1 + kA];
        a.y = ldsA[col * K1 + kA + 1];
        b.x = ldsHR[kA * MPAD + col];
        b.y = ldsHR[(kA + 1) * MPAD + col];
        acc3 = __builtin_amdgcn_wmma_f32_16x16x4_f32(false, a, false, b,
                                                     (short)0, acc3, false, false);
    }
    // + h @ root2 + b2
#pragma unroll
    for (int v = 0; v < 8; ++v) {
        int n = v + 8 * half;
        if (n < NN && col < CC) {
            float hr = 0.0f;
#pragma unroll
            for (int h = 0; h < HH; ++h) hr += ldsH[n * HH + h] * root2[h * CC + col];
            ldsOut[n * CC + col] = acc3[v] + hr + b2[col];
        }
    }
    __syncthreads();

    // log_softmax over C=8, one node per thread
    if (tid < NN) {
        float x[CC];
        float m = -1e30f;
        for (int c = 0; c < CC; ++c) { x[c] = ldsOut[tid * CC + c]; m = fmaxf(m, x[c]); }
        float s = 0.0f;
        for (int c = 0; c < CC; ++c) s += __expf(x[c] - m);
        float lse = __logf(s);
        for (int c = 0; c < CC; ++c) out[tid * CC + c] = x[c] - m - lse;
    }
}

extern "C" void kernel_launch(void* const* d_in, const int* in_sizes, int n_in,
                              void* d_out, int out_size, void* d_ws, size_t ws_size,
                              hipStream_t stream) {
    // setup_inputs order: x, edge_index, edge_type, W1, root1, b1, W2, root2, b2
    const int*   ei    = (const int*)d_in[1];
    const int*   et    = (const int*)d_in[2];
    const float* W1    = (const float*)d_in[3];
    const float* root1 = (const float*)d_in[4];
    const float* b1    = (const float*)d_in[5];
    const float* W2    = (const float*)d_in[6];
    const float* root2 = (const float*)d_in[7];
    const float* b2    = (const float*)d_in[8];
    unsigned int* cnt3 = (unsigned int*)d_ws;   // 3136 u32 scratch bins
    const int E = in_sizes[2];

    rgcn_zero<<<1, 256, 0, stream>>>(cnt3);
    rgcn_hist<<<768, 256, 0, stream>>>(ei, et, cnt3, E);
    rgcn_finalize<<<1, 32, 0, stream>>>(cnt3, W1, root1, b1, W2, root2, b2, (float*)d_out);
}